// GCN_DAE_28295244546757
// MI455X (gfx1250) — compile-verified
//
#include <hip/hip_runtime.h>

// ---------------------------------------------------------------------------
// GCN-DAE forward for MI455X (gfx1250), wave32.
//  - All GEMMs: v_wmma_f32_16x16x4_f32, 128x64 block tile, 8 waves, 2x2 WMMA
//    tiles per wave, double-buffered LDS staging.
//  - Tile staging uses GLOBAL_LOAD_ASYNC_TO_LDS_B128 (ASYNCcnt pipeline) when
//    the toolchain exposes the builtin; falls back to sync v4f ld/st.
// ---------------------------------------------------------------------------

typedef __attribute__((ext_vector_type(2))) float v2f;
typedef __attribute__((ext_vector_type(4))) float v4f;
typedef __attribute__((ext_vector_type(8))) float v8f;
typedef int b128_t __attribute__((vector_size(16)));  // matches builtin param type

#ifndef __has_builtin
#define __has_builtin(x) 0
#endif

#if defined(__AMDGCN__) && __has_builtin(__builtin_amdgcn_global_load_async_to_lds_b128)
#define HAVE_ASYNC 1
#else
#define HAVE_ASYNC 0
#endif

#define AS1 __attribute__((address_space(1)))
#define AS3 __attribute__((address_space(3)))

// Copy one 16-byte chunk global -> LDS (async when available).
__device__ __forceinline__ void cp16(float* dst, const float* src) {
#if HAVE_ASYNC
  __builtin_amdgcn_global_load_async_to_lds_b128(
      (AS1 b128_t*)(AS1 void*)(float*)src,
      (AS3 b128_t*)(AS3 void*)dst, 0, 0);
#else
  *(v4f*)dst = *(const v4f*)src;
#endif
}

#if HAVE_ASYNC
#if __has_builtin(__builtin_amdgcn_s_wait_asynccnt)
#define WAIT_ASYNC(n) __builtin_amdgcn_s_wait_asynccnt(n)
#else
#define WAIT_ASYNC(n) asm volatile("s_wait_asynccnt %0" ::"i"(n) : "memory")
#endif
#else
#define WAIT_ASYNC(n) ((void)0)
#endif

#define BM 128
#define BN 64
#define BK 16
#define LDA 20   // As[m][k] padded row stride (floats): 80B rows, 16B aligned
#define LDBT 20  // Bs[n][k] stride for TRANS_B
#define LDBN 68  // Bs[k][n] stride for normal B: 272B rows, 16B aligned
#define BSZ 1280 // floats per B stage buffer: max(64*20, 16*68)

// C[M x Ncols] = epilogue( A[M x K] @ B + bias )
//   TRANS_B:  B is [Ncols x K] row-major, used as B^T (emb @ emb^T)
//   GCN:      x = rowScale[r] * (acc + Yadd[r,c])  (identity hop + D^-1/2)
//   STREAM_C: nontemporal C stores (for the 256MB sim matrix)
// Requires M % 128 == 0, Ncols % 64 == 0, K % 16 == 0 (true for all uses).
template <bool TRANS_B, bool HAS_BIAS, bool RELU, bool GCN, bool STREAM_C>
__global__ __launch_bounds__(256) void gemm_wmma_f32(
    const float* __restrict__ A, const float* __restrict__ B,
    const float* __restrict__ bias, const float* __restrict__ Yadd,
    const float* __restrict__ rowScale, float* __restrict__ C,
    int M, int Ncols, int K) {
  (void)M;
  __shared__ float As[2][BM * LDA];
  __shared__ float Bs[2][BSZ];

  const int tid  = threadIdx.x;
  const int lane = tid & 31;
  const int wave = tid >> 5;
  const int lm   = lane & 15;  // row/col within 16-tile
  const int lk   = lane >> 4;  // k-half selector
  const int wm   = wave & 3;   // 4 waves along M (32 rows each)
  const int wn   = wave >> 2;  // 2 waves along N (32 cols each)
  const int blockM = blockIdx.y * BM;
  const int blockN = blockIdx.x * BN;

  // ---- staging coordinates (hoisted; pointer-stepped per tile) ----
  const int ar  = tid >> 2;         // 0..63
  const int ac4 = (tid & 3) * 4;    // 0,4,8,12
  const float* aSrc0 = A + (size_t)(blockM + ar) * K + ac4;
  const float* aSrc1 = aSrc0 + (size_t)64 * K;
  const int aDst0 = ar * LDA + ac4;
  const int aDst1 = (ar + 64) * LDA + ac4;

  const float* bSrc;
  int bDst;
  if (TRANS_B) {
    const int bn  = tid >> 2;
    const int bc4 = (tid & 3) * 4;
    bSrc = B + (size_t)(blockN + bn) * K + bc4;
    bDst = bn * LDBT + bc4;
  } else {
    const int bk  = tid >> 4;        // 0..15
    const int bn4 = (tid & 15) * 4;  // 0..60
    bSrc = B + (size_t)bk * Ncols + blockN + bn4;
    bDst = bk * LDBN + bn4;
  }
  const size_t bStep = TRANS_B ? (size_t)BK : (size_t)BK * Ncols;

  v8f acc[2][2] = {};
  const int T = K / BK;

  // prologue: stage tile 0 into buffer 0
  cp16(&As[0][aDst0], aSrc0);
  cp16(&As[0][aDst1], aSrc1);
  cp16(&Bs[0][bDst], bSrc);

  for (int i = 0; i < T; ++i) {
    const int cur = i & 1;
    if (i + 1 < T) {  // prefetch next tile into the other buffer
      const int nxt = cur ^ 1;
      const size_t ka = (size_t)(i + 1) * BK;
      cp16(&As[nxt][aDst0], aSrc0 + ka);
      cp16(&As[nxt][aDst1], aSrc1 + ka);
      cp16(&Bs[nxt][bDst], bSrc + (size_t)(i + 1) * bStep);
      WAIT_ASYNC(3);  // <=3 outstanding -> current tile has landed
    } else {
      WAIT_ASYNC(0);
    }
    __syncthreads();

    const float* as = As[cur];
    const float* bs = Bs[cur];
#pragma unroll
    for (int ks = 0; ks < 4; ++ks) {
      const int kb = ks * 4 + lk * 2;  // per-lane k pair within BK step
      v2f a0 = *(const v2f*)&as[(wm * 32 + lm) * LDA + kb];
      v2f a1 = *(const v2f*)&as[(wm * 32 + 16 + lm) * LDA + kb];
      v2f b0, b1;
      if (TRANS_B) {
        b0 = *(const v2f*)&bs[(wn * 32 + lm) * LDBT + kb];
        b1 = *(const v2f*)&bs[(wn * 32 + 16 + lm) * LDBT + kb];
      } else {
        const int n0 = wn * 32 + lm;
        b0.x = bs[kb * LDBN + n0];
        b0.y = bs[(kb + 1) * LDBN + n0];
        b1.x = bs[kb * LDBN + n0 + 16];
        b1.y = bs[(kb + 1) * LDBN + n0 + 16];
      }
      acc[0][0] = __builtin_amdgcn_wmma_f32_16x16x4_f32(false, a0, false, b0, (short)0, acc[0][0], false, false);
      acc[0][1] = __builtin_amdgcn_wmma_f32_16x16x4_f32(false, a0, false, b1, (short)0, acc[0][1], false, false);
      acc[1][0] = __builtin_amdgcn_wmma_f32_16x16x4_f32(false, a1, false, b0, (short)0, acc[1][0], false, false);
      acc[1][1] = __builtin_amdgcn_wmma_f32_16x16x4_f32(false, a1, false, b1, (short)0, acc[1][1], false, false);
    }
    __syncthreads();
  }

  // ---- epilogue (C 16x16 layout: VGPR v -> row 8*lk+v, col lm) ----
#pragma unroll
  for (int mi = 0; mi < 2; ++mi)
#pragma unroll
    for (int ni = 0; ni < 2; ++ni) {
#pragma unroll
      for (int v = 0; v < 8; ++v) {
        const int r = blockM + wm * 32 + mi * 16 + lk * 8 + v;
        const int c = blockN + wn * 32 + ni * 16 + lm;
        float x = acc[mi][ni][v];
        if (HAS_BIAS) x += bias[c];
        if (GCN) x = rowScale[r] * (x + Yadd[(size_t)r * Ncols + c]);
        if (RELU) x = fmaxf(x, 0.0f);
        if (STREAM_C)
          __builtin_nontemporal_store(x, &C[(size_t)r * Ncols + c]);
        else
          C[(size_t)r * Ncols + c] = x;
      }
    }
}

// Row-wise L2 normalize in place: x[i,:] /= max(||x[i,:]||, 1e-12)
__global__ __launch_bounds__(256) void normalize_rows(float* __restrict__ x, int D) {
  __shared__ float red[8];
  const int row = blockIdx.x;
  float* p = x + (size_t)row * D;
  float s = 0.0f;
  for (int j = threadIdx.x; j < D; j += 256) { float v = p[j]; s += v * v; }
#pragma unroll
  for (int off = 16; off > 0; off >>= 1) s += __shfl_down(s, off);
  if ((threadIdx.x & 31) == 0) red[threadIdx.x >> 5] = s;
  __syncthreads();
  if (threadIdx.x == 0) {
    float t = 0.0f;
#pragma unroll
    for (int i = 0; i < 8; ++i) t += red[i];
    red[0] = 1.0f / fmaxf(sqrtf(t), 1e-12f);
  }
  __syncthreads();
  const float inv = red[0];
  for (int j = threadIdx.x; j < D; j += 256) p[j] *= inv;
}

// Per-row: keep top-K entries (threshold = K-th max), zero diagonal, relu. In place.
__global__ __launch_bounds__(256) void topk_mask(float* __restrict__ Mm, int n, int K) {
  __shared__ float vals[8192];
  __shared__ float red[256];
  const int row = blockIdx.x;
  float* p = Mm + (size_t)row * n;
  for (int j = threadIdx.x; j < n; j += 256) vals[j] = __builtin_nontemporal_load(&p[j]);
  __syncthreads();

  float thresh = 0.0f;
  for (int it = 0; it < K; ++it) {
    float m = -3.402823466e38f;
    for (int j = threadIdx.x; j < n; j += 256) m = fmaxf(m, vals[j]);
    red[threadIdx.x] = m;
    __syncthreads();
    for (int s = 128; s > 0; s >>= 1) {
      if (threadIdx.x < s) red[threadIdx.x] = fmaxf(red[threadIdx.x], red[threadIdx.x + s]);
      __syncthreads();
    }
    const float gmax = red[0];
    __syncthreads();
    if (it == K - 1) {
      thresh = gmax;
    } else {
      for (int j = threadIdx.x; j < n; j += 256)
        if (vals[j] == gmax) vals[j] = -3.402823466e38f;
      __syncthreads();
    }
  }
  for (int j = threadIdx.x; j < n; j += 256) {
    const float v = p[j];
    p[j] = (v >= thresh && j != row) ? fmaxf(v, 0.0f) : 0.0f;
  }
}

// out[i,j] = 0.5*(M[i,j] + M[j,i]) with LDS tile transpose (both reads coalesced)
__global__ __launch_bounds__(1024) void symmetrize(const float* __restrict__ Mm,
                                                   float* __restrict__ out, int n) {
  __shared__ float t[32][33];
  const int bi = blockIdx.y, bj = blockIdx.x;
  const int tx = threadIdx.x, ty = threadIdx.y;
  t[ty][tx] = Mm[(size_t)(bj * 32 + ty) * n + bi * 32 + tx];
  __syncthreads();
  const size_t idx = (size_t)(bi * 32 + ty) * n + bj * 32 + tx;
  out[idx] = 0.5f * (Mm[idx] + t[tx][ty]);
}

// dis[i] = 1 / (sqrt(1 + sum_j adj[i,j]) + 1e-10)
__global__ __launch_bounds__(256) void degree_scale(const float* __restrict__ adj,
                                                    float* __restrict__ dis, int n) {
  __shared__ float red[256];
  const int row = blockIdx.x;
  const float* p = adj + (size_t)row * n;
  float s = 0.0f;
  for (int j = threadIdx.x; j < n; j += 256) s += p[j];
  red[threadIdx.x] = s;
  __syncthreads();
  for (int st = 128; st > 0; st >>= 1) {
    if (threadIdx.x < st) red[threadIdx.x] += red[threadIdx.x + st];
    __syncthreads();
  }
  if (threadIdx.x == 0) dis[row] = 1.0f / (sqrtf(red[0] + 1.0f) + 1e-10f);
}

// x[i,:] *= dis[i]  (cols must be a multiple of 256)
__global__ __launch_bounds__(256) void rowscale(float* __restrict__ x,
                                                const float* __restrict__ dis, int cols) {
  const int row = blockIdx.y;
  const int j = blockIdx.x * 256 + threadIdx.x;
  x[(size_t)row * cols + j] *= dis[row];
}

extern "C" void kernel_launch(void* const* d_in, const int* in_sizes, int n_in,
                              void* d_out, int out_size, void* d_ws, size_t ws_size,
                              hipStream_t stream) {
  (void)in_sizes; (void)n_in; (void)out_size; (void)ws_size;
  const float* features = (const float*)d_in[0];
  const float* masked   = (const float*)d_in[1];
  const float* W1  = (const float*)d_in[2];
  const float* b1  = (const float*)d_in[3];
  const float* W2  = (const float*)d_in[4];
  const float* b2  = (const float*)d_in[5];
  const float* Wg1 = (const float*)d_in[6];
  const float* bg1 = (const float*)d_in[7];
  const float* Wg2 = (const float*)d_in[8];
  const float* bg2 = (const float*)d_in[9];

  const int R = 8192, D = 512, H = 256, KTOP = 21;

  float* out = (float*)d_out;
  float* h3  = out;                    // [R, D]
  float* adj = out + (size_t)R * D;    // [R, R]  gen_adj output; also GEMM operand

  float* Msim = (float*)d_ws;                  // [R, R] sim -> masked sim
  float* bufA = Msim + (size_t)R * R;          // [R, D] h   -> h1 ([R,H] fits)
  float* bufB = bufA + (size_t)R * D;          // [R, D] emb -> x2/y2
  float* bufC = bufB + (size_t)R * D;          // [R, H] xw1 -> y1 (in place)
  float* dis  = bufC + (size_t)R * H;          // [R]

  const dim3 blk(256);

  // 1) h = relu(features @ W1 + b1)
  gemm_wmma_f32<false, true, true, false, false><<<dim3(D / BN, R / BM), blk, 0, stream>>>(
      features, W1, b1, nullptr, nullptr, bufA, R, D, D);
  // 2) emb_raw = h @ W2 + b2
  gemm_wmma_f32<false, true, false, false, false><<<dim3(D / BN, R / BM), blk, 0, stream>>>(
      bufA, W2, b2, nullptr, nullptr, bufB, R, D, D);
  // 3) emb = normalize(emb_raw)
  normalize_rows<<<R, blk, 0, stream>>>(bufB, D);
  // 4) sim = emb @ emb^T  (streaming nontemporal store: 256MB > L2)
  gemm_wmma_f32<true, false, false, false, true><<<dim3(R / BN, R / BM), blk, 0, stream>>>(
      bufB, bufB, nullptr, nullptr, nullptr, Msim, R, R, D);
  // 5) top-k(21) mask + relu + zero diag, in place
  topk_mask<<<R, blk, 0, stream>>>(Msim, R, KTOP);
  // 6) gen_adj = 0.5 * (M + M^T)  -> output region
  symmetrize<<<dim3(R / 32, R / 32), dim3(32, 32), 0, stream>>>(Msim, adj, R);
  // 7) dis[i] = 1/(sqrt(1 + rowsum(gen_adj)) + eps)   [A = gen_adj + I, never materialized]
  degree_scale<<<R, blk, 0, stream>>>(adj, dis, R);
  // 8) xw1 = masked @ Wg1 + bg1
  gemm_wmma_f32<false, true, false, false, false><<<dim3(H / BN, R / BM), blk, 0, stream>>>(
      masked, Wg1, bg1, nullptr, nullptr, bufC, R, H, D);
  // 9) y1 = dis ⊙ xw1 (in place)
  rowscale<<<dim3(H / 256, R), blk, 0, stream>>>(bufC, dis, H);
  // 10) h1 = relu(dis ⊙ (gen_adj @ y1 + y1))
  gemm_wmma_f32<false, false, true, true, false><<<dim3(H / BN, R / BM), blk, 0, stream>>>(
      adj, bufC, nullptr, bufC, dis, bufA, R, H, R);
  // 11) x2 = h1 @ Wg2 + bg2
  gemm_wmma_f32<false, true, false, false, false><<<dim3(D / BN, R / BM), blk, 0, stream>>>(
      bufA, Wg2, bg2, nullptr, nullptr, bufB, R, D, H);
  // 12) y2 = dis ⊙ x2 (in place)
  rowscale<<<dim3(D / 256, R), blk, 0, stream>>>(bufB, dis, D);
  // 13) h3 = dis ⊙ (gen_adj @ y2 + y2)  -> output region
  gemm_wmma_f32<false, false, false, true, false><<<dim3(D / BN, R / BM), blk, 0, stream>>>(
      adj, bufB, nullptr, bufB, dis, h3, R, D, R);
}